// get_loss_13829794693841
// MI455X (gfx1250) — compile-verified
//
#include <hip/hip_runtime.h>
#include <hip/hip_bf16.h>
#include <math.h>

#define B_ 8
#define N_ 131072
#define C_ 20
#define K_ 20

typedef __attribute__((ext_vector_type(2))) float v2f;
typedef __attribute__((ext_vector_type(8))) float v8f;

// ---------------------------------------------------------------------------
// Kernel 0: zero the workspace accumulators (must run every call; the harness
// does not re-poison between graph replays).
// ---------------------------------------------------------------------------
__global__ void init_ws_kernel(float* __restrict__ ws, int n) {
    int i = blockIdx.x * blockDim.x + threadIdx.x;
    if (i < n) ws[i] = 0.0f;
}

// ---------------------------------------------------------------------------
// Kernel 1: per-batch segment sums via V_WMMA_F32_16X16X4_F32.
//   D[m][c] += sum_k onehot(inst[base+k]==m) * val(k,c)
//   val(k,0..2) = point coords, val(k,3) = 1.0 (count), cols 4..15 = 0.
// A (16x4 f32, 2 VGPR):  lane<16: rows M=lane, K=0/1 ; lane>=16: K=2/3
// B (4x16 f32, 2 VGPR):  VGPR j: K = j + 2*(lane>>4), N = lane&15
// C (16x16 f32, 8 VGPR): VGPR j: M = j + 8*(lane>>4), N = lane&15
// Two accumulators cover segments 0-15 and 16-19 (K_=20).
// ---------------------------------------------------------------------------
__global__ __launch_bounds__(256) void seg_sum_wmma_kernel(
        const float* __restrict__ points,   // [B][N][3]
        const int*   __restrict__ label,    // [B][2][N]
        float*       __restrict__ gacc)     // [B][K][4] (x,y,z,cnt)
{
    __shared__ float lacc[K_ * 4];
    const int b   = blockIdx.y;
    const int tid = threadIdx.x;
    if (tid < K_ * 4) lacc[tid] = 0.0f;
    __syncthreads();

    const int lane = tid & 31;
    const int wave = tid >> 5;
    const int half = lane >> 4;   // 0: lanes 0-15, 1: lanes 16-31
    const int idx  = lane & 15;   // row (A) / column (B,C) index

    const int wavesPerBlock = blockDim.x >> 5;
    const int gwave         = blockIdx.x * wavesPerBlock + wave;
    const int waveStride    = gridDim.x * wavesPerBlock;

    const int*   inst = label  + (size_t)b * 2 * N_ + N_;  // instance row
    const float* pts  = points + (size_t)b * N_ * 3;

    v8f c_lo = {};   // segments 0..15
    v8f c_hi = {};   // segments 16..31 (only 16..19 ever hit)

    const int ngroups = N_ / 4;  // 4 points (K=4) per WMMA pair
    for (int g = gwave; g < ngroups; g += waveStride) {
        const int base = g * 4;
        const int k0   = base + 2 * half;
        const int k1   = k0 + 1;
        const int id0  = inst[k0];
        const int id1  = inst[k1];

        v2f a_lo, a_hi, bm;
        a_lo.x = (id0 == idx)        ? 1.0f : 0.0f;
        a_lo.y = (id1 == idx)        ? 1.0f : 0.0f;
        a_hi.x = (id0 == (idx + 16)) ? 1.0f : 0.0f;
        a_hi.y = (id1 == (idx + 16)) ? 1.0f : 0.0f;

        float b0, b1;
        if (idx < 3) {            // x / y / z column
            b0 = pts[(size_t)k0 * 3 + idx];
            b1 = pts[(size_t)k1 * 3 + idx];
        } else if (idx == 3) {    // count column
            b0 = 1.0f; b1 = 1.0f;
        } else {                  // dead columns
            b0 = 0.0f; b1 = 0.0f;
        }
        bm.x = b0; bm.y = b1;

        // 8 args: (neg_a, A, neg_b, B, c_mod, C, reuse_a, reuse_b)
        c_lo = __builtin_amdgcn_wmma_f32_16x16x4_f32(
                   false, a_lo, false, bm, (short)0, c_lo, false, false);
        c_hi = __builtin_amdgcn_wmma_f32_16x16x4_f32(
                   false, a_hi, false, bm, (short)0, c_hi, false, false);
    }

    // Fold wave accumulators into the block-level LDS accumulator.
    if (idx < 4) {
        #pragma unroll
        for (int j = 0; j < 8; ++j) {
            const int seg = j + 8 * half;         // 0..15
            atomicAdd(&lacc[seg * 4 + idx], c_lo[j]);
        }
        if (half == 0) {
            #pragma unroll
            for (int j = 0; j < 4; ++j) {         // segments 16..19
                atomicAdd(&lacc[(16 + j) * 4 + idx], c_hi[j]);
            }
        }
    }
    __syncthreads();
    if (tid < K_ * 4) atomicAdd(&gacc[b * (K_ * 4) + tid], lacc[tid]);
}

// ---------------------------------------------------------------------------
// Kernel 2: centers = sum / max(count, 1)
// ---------------------------------------------------------------------------
__global__ void compute_centers_kernel(const float* __restrict__ gacc,
                                       float* __restrict__ centers) {
    int i = blockIdx.x * blockDim.x + threadIdx.x;   // over B*K
    if (i < B_ * K_) {
        const float cnt = gacc[i * 4 + 3];
        const float inv = 1.0f / fmaxf(cnt, 1.0f);
        centers[i * 4 + 0] = gacc[i * 4 + 0] * inv;
        centers[i * 4 + 1] = gacc[i * 4 + 1] * inv;
        centers[i * 4 + 2] = gacc[i * 4 + 2] * inv;
        centers[i * 4 + 3] = 0.0f;
    }
}

// ---------------------------------------------------------------------------
// Kernel 3: fused sem-loss gather + center-shift loss, global reduction.
// ---------------------------------------------------------------------------
__global__ __launch_bounds__(256) void loss_kernel(
        const float* __restrict__ points,    // [B][N][3]
        const float* __restrict__ sem,       // [B][N][C]
        const float* __restrict__ csv,       // [B][N][3]
        const int*   __restrict__ label,     // [B][2][N]
        const float* __restrict__ centers,   // [B][K][4]
        float*       __restrict__ sums)      // [2] : sem_sum, csv_sum
{
    float sem_acc = 0.0f, csv_acc = 0.0f;
    const long long total  = (long long)B_ * N_;
    const long long stride = (long long)gridDim.x * blockDim.x;

    for (long long i = (long long)blockIdx.x * blockDim.x + threadIdx.x;
         i < total; i += stride) {
        const int b = (int)(i >> 17);        // N_ == 2^17
        const int n = (int)(i & (N_ - 1));

        if (i + stride < total)              // strided gather: prefetch next row
            __builtin_prefetch(&sem[(i + stride) * C_], 0, 1);

        const int cls  = label[(size_t)b * 2 * N_ + n];
        const int inst = label[(size_t)b * 2 * N_ + N_ + n];

        sem_acc += sem[i * C_ + cls];

        const float px = points[i * 3 + 0];
        const float py = points[i * 3 + 1];
        const float pz = points[i * 3 + 2];
        const float* ctr = centers + ((size_t)b * K_ + inst) * 4;
        const float dx = ctr[0] - px;
        const float dy = ctr[1] - py;
        const float dz = ctr[2] - pz;
        const float w  = fminf(sqrtf(dx * dx + dy * dy + dz * dz), 1.0f);
        const float ex = dx - csv[i * 3 + 0];
        const float ey = dy - csv[i * 3 + 1];
        const float ez = dz - csv[i * 3 + 2];
        csv_acc += sqrtf(ex * ex + ey * ey + ez * ez) * w;
    }

    // wave32 reduction
    #pragma unroll
    for (int off = 16; off > 0; off >>= 1) {
        sem_acc += __shfl_xor(sem_acc, off, 32);
        csv_acc += __shfl_xor(csv_acc, off, 32);
    }
    __shared__ float s0[8], s1[8];
    const int wave = threadIdx.x >> 5;
    const int lane = threadIdx.x & 31;
    if (lane == 0) { s0[wave] = sem_acc; s1[wave] = csv_acc; }
    __syncthreads();
    if (wave == 0) {
        const int nw = blockDim.x >> 5;
        float a = (lane < nw) ? s0[lane] : 0.0f;
        float c = (lane < nw) ? s1[lane] : 0.0f;
        #pragma unroll
        for (int off = 4; off > 0; off >>= 1) {
            a += __shfl_xor(a, off, 32);
            c += __shfl_xor(c, off, 32);
        }
        if (lane == 0) {
            atomicAdd(&sums[0], a);
            atomicAdd(&sums[1], c);
        }
    }
}

// ---------------------------------------------------------------------------
// Kernel 4: final scalar: loss = -sem_mean + 0.2 * csv_mean
// ---------------------------------------------------------------------------
__global__ void finalize_kernel(const float* __restrict__ sums,
                                float* __restrict__ out) {
    if (blockIdx.x == 0 && threadIdx.x == 0) {
        const float invM = 1.0f / (float)((long long)B_ * N_);
        out[0] = -sums[0] * invM + 0.2f * (sums[1] * invM);
    }
}

// ---------------------------------------------------------------------------
// Workspace layout (floats):
//   [0,    640)  gacc     : B*K*4 segment accumulators (x,y,z,cnt)
//   [640, 1280)  centers  : B*K*4
//   [1280,1282)  sums     : sem_sum, csv_sum
// ---------------------------------------------------------------------------
extern "C" void kernel_launch(void* const* d_in, const int* in_sizes, int n_in,
                              void* d_out, int out_size, void* d_ws, size_t ws_size,
                              hipStream_t stream) {
    const float* points = (const float*)d_in[0];
    const float* sem    = (const float*)d_in[1];
    const float* csv    = (const float*)d_in[2];
    const int*   label  = (const int*)d_in[3];   // harness integer convention

    float* ws      = (float*)d_ws;
    float* gacc    = ws;
    float* centers = ws + 640;
    float* sums    = ws + 1280;

    init_ws_kernel<<<6, 256, 0, stream>>>(ws, 1282);

    dim3 g1(64, B_);
    seg_sum_wmma_kernel<<<g1, 256, 0, stream>>>(points, label, gacc);

    compute_centers_kernel<<<1, 256, 0, stream>>>(gacc, centers);

    loss_kernel<<<2048, 256, 0, stream>>>(points, sem, csv, label, centers, sums);

    finalize_kernel<<<1, 1, 0, stream>>>(sums, (float*)d_out);
}